// GroupedVisitEncoder_5772436045898
// MI455X (gfx1250) — compile-verified
//
#include <hip/hip_runtime.h>
#include <hip/hip_bf16.h>

typedef __attribute__((ext_vector_type(2))) float v2f;
typedef __attribute__((ext_vector_type(4))) float v4f;
typedef __attribute__((ext_vector_type(8))) float v8f;

#define TPB_TOKENS 8
#define HIDDEN 256
#define NGROUP 16
#define NWAVE 8

// branch-free tanh: sign(x)*(1-e^{-2|x|})/(1+e^{-2|x|}) via v_exp_f32 + v_rcp_f32
__device__ __forceinline__ float fast_tanh(float v) {
  const float a = fabsf(v);
  const float e = __expf(-2.0f * a);
  const float r = __builtin_amdgcn_rcpf(1.0f + e);
  const float t = (1.0f - e) * r;
  return copysignf(t, v);
}

// tokens[16][256] per token:  t[g][h] = gv[g]*w_gp[h] + b_gp[h]
// scores[g] = sum_h tanh(t[g][h]) * w_sp[h] + b_sp ; softmax -> weights
// visit[h]  = (sum_g weights[g]*gv[g]) * w_gp[h] + b_gp[h]   (weights sum to 1)
__global__ __launch_bounds__(256) void gve_kernel(
    const float* __restrict__ x,      // (ntok, 64)
    const float* __restrict__ w_gp,   // (256,)
    const float* __restrict__ b_gp,   // (256,)
    const float* __restrict__ w_sp,   // (256,)
    const float* __restrict__ b_sp,   // (1,)
    float* __restrict__ visit_out,    // (ntok, 256)
    float* __restrict__ tokens_out,   // (ntok, 16, 256)
    float* __restrict__ weights_out,  // (ntok, 16)
    int ntok)
{
  __shared__ float wg[HIDDEN];
  __shared__ float bg[HIDDEN];
  __shared__ float ws[HIDDEN];
  __shared__ float gv[NGROUP];
  __shared__ float scp[NWAVE][NGROUP];  // per-wave partial scores
  __shared__ float s_tok;

  const int tid = threadIdx.x;
  const int wv = tid >> 5;        // wave id 0..7
  const int lane = tid & 31;

  // stage parameter vectors in LDS (once per block)
  wg[tid] = w_gp[tid];
  bg[tid] = b_gp[tid];
  ws[tid] = w_sp[tid];
  const float bsp = b_sp[0];

  // per-thread register slice of w_gp/b_gp for the tokens-store phase:
  // h-slice is invariant across iterations (i*1024 mod 256 == 0)
  const int h0s = (tid * 4) & (HIDDEN - 1);
  const v4f wgr = *(const v4f*)(w_gp + h0s);
  const v4f bgr = *(const v4f*)(b_gp + h0s);

  __syncthreads();

  const int tk0 = blockIdx.x * TPB_TOKENS;
  for (int it = 0; it < TPB_TOKENS; ++it) {
    const int tk = tk0 + it;
    if (tk >= ntok) break;

    // ---- phase 1: segment means gv[16] (threads 0..15) ----
    if (tid < NGROUP) {
      const int cumA[4] = {0, 1, 4, 9};
      const int cszA[4] = {1, 3, 5, 7};
      const int q = tid >> 2, r = tid & 3;
      const int start = q * 16 + cumA[r];
      const int cnt = cszA[r];
      const float* xp = x + (size_t)tk * 64 + start;
      float s = 0.0f;
      for (int j = 0; j < cnt; ++j) s += xp[j];
      gv[tid] = s / (float)cnt;
    }
    __syncthreads();

    // ---- phase 2 (all 8 waves): stream tokens tile, NT b128 stores ----
    {
      float* tbase = tokens_out + (size_t)tk * (NGROUP * HIDDEN);
      #pragma unroll
      for (int i = 0; i < 4; ++i) {
        const int g = i * 4 + (tid >> 6);
        const float gvg = gv[g];
        v4f v;
        v.x = fmaf(gvg, wgr.x, bgr.x);
        v.y = fmaf(gvg, wgr.y, bgr.y);
        v.z = fmaf(gvg, wgr.z, bgr.z);
        v.w = fmaf(gvg, wgr.w, bgr.w);
        __builtin_nontemporal_store(v, (v4f*)(tbase + i * 1024 + tid * 4));
      }
    }

    // ---- phase 3 (ALL waves, full EXEC): scores via V_WMMA_F32_16X16X4_F32
    // Wave w accumulates K-chunks c in [8w, 8w+8).
    // A[g][k] = tanh(token)*w_sp (A layout: lanes0-15 K=0,1; lanes16-31 K=2,3)
    // B = all-ones  =>  D[g][n] = sum_k A[g][k]  (B layout-independent)
    {
      const int g = lane & 15;
      const int hi = lane >> 4;
      const float gvg = gv[g];
      v8f acc = {};
      v2f one;
      one.x = 1.0f;
      one.y = 1.0f;
      #pragma unroll
      for (int cc = 0; cc < 8; ++cc) {
        const int h0 = (wv * 8 + cc) * 4 + 2 * hi;
        v2f a;
        a.x = fast_tanh(fmaf(gvg, wg[h0],     bg[h0]))     * ws[h0];
        a.y = fast_tanh(fmaf(gvg, wg[h0 + 1], bg[h0 + 1])) * ws[h0 + 1];
        acc = __builtin_amdgcn_wmma_f32_16x16x4_f32(
            false, a, false, one, (short)0, acc, false, false);
      }
      // C/D layout: VGPR r -> (M=r, N=lane) lanes0-15 ; (M=8+r, N=lane-16) lanes16-31
      if (lane == 0) {
        #pragma unroll
        for (int r = 0; r < 8; ++r) scp[wv][r] = acc[r];
      } else if (lane == 16) {
        #pragma unroll
        for (int r = 0; r < 8; ++r) scp[wv][8 + r] = acc[r];
      }
    }
    __syncthreads();

    // ---- phase 4 (lanes 0..15 of wave 0): combine partials, softmax ----
    if (tid < NGROUP) {
      float s = bsp;
      #pragma unroll
      for (int w = 0; w < NWAVE; ++w) s += scp[w][tid];
      float m = s;
      #pragma unroll
      for (int off = 8; off > 0; off >>= 1) m = fmaxf(m, __shfl_xor(m, off));
      const float e = __expf(s - m);
      float su = e;
      #pragma unroll
      for (int off = 8; off > 0; off >>= 1) su += __shfl_xor(su, off);
      const float w = e / su;
      weights_out[(size_t)tk * NGROUP + tid] = w;
      float p = w * gv[tid];
      #pragma unroll
      for (int off = 8; off > 0; off >>= 1) p += __shfl_xor(p, off);
      if (tid == 0) s_tok = p;
    }
    __syncthreads();

    // ---- phase 5: visit[h] = s*w_gp[h] + b_gp[h] ----
    {
      const float sv = s_tok;
      __builtin_nontemporal_store(fmaf(sv, wg[tid], bg[tid]),
                                  visit_out + (size_t)tk * HIDDEN + tid);
    }
    __syncthreads();  // protect gv/scp/s_tok before next token
  }
}

extern "C" void kernel_launch(void* const* d_in, const int* in_sizes, int n_in,
                              void* d_out, int out_size, void* d_ws, size_t ws_size,
                              hipStream_t stream) {
  const float* x    = (const float*)d_in[0];
  const float* w_gp = (const float*)d_in[1];
  const float* b_gp = (const float*)d_in[2];
  const float* w_sp = (const float*)d_in[3];
  const float* b_sp = (const float*)d_in[4];

  const int ntok = in_sizes[0] / 64;  // 32*2048 = 65536

  float* out = (float*)d_out;
  float* visit_out   = out;
  float* tokens_out  = visit_out + (size_t)ntok * HIDDEN;
  float* weights_out = tokens_out + (size_t)ntok * NGROUP * HIDDEN;

  const int grid = (ntok + TPB_TOKENS - 1) / TPB_TOKENS;
  gve_kernel<<<grid, 256, 0, stream>>>(x, w_gp, b_gp, w_sp, b_sp,
                                       visit_out, tokens_out, weights_out, ntok);
}